// VQVAE_57793079935368
// MI455X (gfx1250) — compile-verified
//
#include <hip/hip_runtime.h>
#include <hip/hip_bf16.h>

typedef float v2f __attribute__((ext_vector_type(2)));
typedef float v8f __attribute__((ext_vector_type(8)));

#define EPS_BN 1e-5f
#define BETA_VQ 0.2f

__device__ __forceinline__ float wave_sum(float v) {
  #pragma unroll
  for (int o = 16; o > 0; o >>= 1) v += __shfl_down(v, o, 32);
  return v;
}

__global__ void k_zero(float* p, int n) {
  int t = blockIdx.x * blockDim.x + threadIdx.x;
  if (t < n) p[t] = 0.f;
}

// conv1: [128,1,256,256] -> [128,16,128,128], k=4 s=2 p=1, raw output + BN stats
__global__ void k_conv1(const float* __restrict__ x, const float* __restrict__ w,
                        const float* __restrict__ b, float* __restrict__ out,
                        float* __restrict__ ssum, float* __restrict__ ssq) {
  __shared__ float sw[256];
  __shared__ float sb[16];
  int tid = threadIdx.x;
  sw[tid] = w[tid];
  if (tid < 16) sb[tid] = b[tid];
  __syncthreads();
  int t = blockIdx.x * 256 + tid;           // 2,097,152 threads (exact)
  int n = t >> 14, yx = t & 16383;
  int oy = yx >> 7, ox = yx & 127;
  int iy0 = 2 * oy - 1, ix0 = 2 * ox - 1;
  const float* xp = x + (size_t)n * 65536;
  float v[16];
  #pragma unroll
  for (int ky = 0; ky < 4; ++ky) {
    int iy = iy0 + ky;
    #pragma unroll
    for (int kx = 0; kx < 4; ++kx) {
      int ix = ix0 + kx;
      bool ok = (iy >= 0) & (iy < 256) & (ix >= 0) & (ix < 256);
      v[ky * 4 + kx] = ok ? xp[iy * 256 + ix] : 0.f;
    }
  }
  #pragma unroll
  for (int oc = 0; oc < 16; ++oc) {
    float a = sb[oc];
    #pragma unroll
    for (int k = 0; k < 16; ++k) a += sw[oc * 16 + k] * v[k];
    out[(((size_t)n * 16 + oc) << 14) + yx] = a;
    float sA = wave_sum(a), sQ = wave_sum(a * a);
    if ((tid & 31) == 0) { atomicAdd(&ssum[oc], sA); atomicAdd(&ssq[oc], sQ); }
  }
}

// fold BN(batch stats) into per-channel scale/shift
__global__ void k_bnfold(const float* ssum, const float* ssq, const float* gam,
                         const float* bet, float* scale, float* shift, int C, float invN) {
  int c = threadIdx.x;
  if (c < C) {
    float m = ssum[c] * invN;
    float var = ssq[c] * invN - m * m;
    float sc = gam[c] * rsqrtf(var + EPS_BN);
    scale[c] = sc;
    shift[c] = bet[c] - m * sc;
  }
}

// conv2: fused bn1+relu on reads; [128,16,128,128] -> [128,4,64,64] raw + stats
__global__ void k_conv2(const float* __restrict__ h1, const float* __restrict__ w,
                        const float* __restrict__ b, const float* __restrict__ sc1,
                        const float* __restrict__ sh1, float* __restrict__ out,
                        float* ssum, float* ssq) {
  __shared__ float sw[1024];
  __shared__ float sb[4], ssc[16], ssh[16];
  int tid = threadIdx.x;
  for (int i = tid; i < 1024; i += 256) sw[i] = w[i];
  if (tid < 4) sb[tid] = b[tid];
  if (tid < 16) { ssc[tid] = sc1[tid]; ssh[tid] = sh1[tid]; }
  __syncthreads();
  int t = blockIdx.x * 256 + tid;           // 524,288 threads (exact)
  int n = t >> 12, yx = t & 4095;
  int oy = yx >> 6, ox = yx & 63;
  int iy0 = 2 * oy - 1, ix0 = 2 * ox - 1;
  float acc[4] = {sb[0], sb[1], sb[2], sb[3]};
  const float* base = h1 + ((size_t)n << 18);
  #pragma unroll
  for (int ic = 0; ic < 16; ++ic) {
    const float* p = base + ((size_t)ic << 14);
    if (ic < 15) __builtin_prefetch(p + (1 << 14) + oy * 256 + ox * 2, 0, 0); // global_prefetch_b8
    float s = ssc[ic], sh = ssh[ic];
    #pragma unroll
    for (int ky = 0; ky < 4; ++ky) {
      int iy = iy0 + ky;
      if (iy < 0 || iy >= 128) continue;
      #pragma unroll
      for (int kx = 0; kx < 4; ++kx) {
        int ix = ix0 + kx;
        if (ix < 0 || ix >= 128) continue;
        float v = fmaxf(0.f, p[iy * 128 + ix] * s + sh);
        #pragma unroll
        for (int oc = 0; oc < 4; ++oc) acc[oc] += v * sw[((oc * 16 + ic) * 4 + ky) * 4 + kx];
      }
    }
  }
  #pragma unroll
  for (int oc = 0; oc < 4; ++oc) {
    out[(((size_t)n * 4 + oc) << 12) + yx] = acc[oc];
    float sA = wave_sum(acc[oc]), sQ = wave_sum(acc[oc] * acc[oc]);
    if ((tid & 31) == 0) { atomicAdd(&ssum[oc], sA); atomicAdd(&ssq[oc], sQ); }
  }
}

// pre 1x1 conv (fused bn2+relu): -> z planar [128,2,64,64]
__global__ void k_pre(const float* __restrict__ h2, const float* __restrict__ pw,
                      const float* __restrict__ pb, const float* sc2, const float* sh2,
                      float* __restrict__ z) {
  int t = blockIdx.x * 256 + threadIdx.x;   // 524,288 pixels
  int n = t >> 12, yx = t & 4095;
  const float* base = h2 + ((size_t)n << 14);
  float z0 = pb[0], z1 = pb[1];
  #pragma unroll
  for (int c = 0; c < 4; ++c) {
    float v = fmaxf(0.f, base[(c << 12) + yx] * sc2[c] + sh2[c]);
    z0 += pw[c] * v;
    z1 += pw[4 + c] * v;
  }
  z[(((size_t)n * 2) << 12) + yx] = z0;
  z[(((size_t)n * 2 + 1) << 12) + yx] = z1;
}

// VQ via V_WMMA_F32_16X16X4_F32: per wave, 16 pixels x 512 codes.
// score = -2 z.e + |e|^2  (|z|^2 is row-constant, irrelevant for argmin).
// A 16x4 layout (ISA 7.12.2): VGPRi lanes0-15 = K=i, lanes16-31 = K=i+2.
//   lanes0-15: {z0, z1}; lanes16-31: {1, 0}  -> K-slots [z0, z1, 1, 0].
// B 4x16 assumed mirrored: VGPRi lanes0-15 = row K=i, lanes16-31 = row K=i+2.
//   lanes0-15: {-2e0, -2e1}; lanes16-31: {|e|^2, 0}.
// D 16x16: VGPR r = row M=r (lanes0-15) / M=r+8 (lanes16-31), N = lane&15.
// Fused: q gather, post 1x1 conv to g, (q-z)^2 loss accumulation.
__global__ void k_vq(const float* __restrict__ z, const float* __restrict__ emb,
                     const float* __restrict__ postw, const float* __restrict__ postb,
                     float* __restrict__ g, float* vq_sum) {
  int tid = threadIdx.x;
  int lane = tid & 31;
  int m = lane & 15;
  int wave = (blockIdx.x * 256 + tid) >> 5;
  int p16 = wave << 4;
  int pm = p16 + m;
  int pn = pm >> 12, pyx = pm & 4095;
  float z0 = z[(((size_t)pn * 2) << 12) + pyx];
  float z1 = z[(((size_t)pn * 2 + 1) << 12) + pyx];
  v2f A;
  if (lane < 16) { A.x = z0; A.y = z1; } else { A.x = 1.0f; A.y = 0.0f; }
  float bestv[8];
  int besti[8];
  #pragma unroll
  for (int r = 0; r < 8; ++r) { bestv[r] = 3.4e38f; besti[r] = 0; }
  for (int t0 = 0; t0 < 32; ++t0) {
    int code = (t0 << 4) + m;
    float e0 = emb[code * 2], e1 = emb[code * 2 + 1];
    v2f B;
    if (lane < 16) { B.x = -2.f * e0; B.y = -2.f * e1; }
    else           { B.x = e0 * e0 + e1 * e1; B.y = 0.f; }
    v8f acc = {0.f, 0.f, 0.f, 0.f, 0.f, 0.f, 0.f, 0.f};
    acc = __builtin_amdgcn_wmma_f32_16x16x4_f32(false, A, false, B, (short)0, acc,
                                                false, false);
    #pragma unroll
    for (int r = 0; r < 8; ++r) {
      float s = acc[r];                       // column = t0*16 + m for this lane
      if (s < bestv[r]) { bestv[r] = s; besti[r] = code; }
    }
  }
  float local = 0.f;
  #pragma unroll
  for (int r = 0; r < 8; ++r) {
    float bv = bestv[r];
    int bi = besti[r];
    #pragma unroll
    for (int off = 8; off >= 1; off >>= 1) {  // xor masks <16: reduce within each half
      float ov = __shfl_xor(bv, off, 32);
      int oi = __shfl_xor(bi, off, 32);
      if (ov < bv || (ov == bv && oi < bi)) { bv = ov; bi = oi; }
    }
    if (m == 0) {                             // lane0 -> M=r, lane16 -> M=r+8
      int M = r + ((lane >> 4) << 3);
      int p = p16 + M;
      int nn = p >> 12, yx = p & 4095;
      float q0 = emb[bi * 2], q1 = emb[bi * 2 + 1];
      float zz0 = z[(((size_t)nn * 2) << 12) + yx];
      float zz1 = z[(((size_t)nn * 2 + 1) << 12) + yx];
      float d0 = q0 - zz0, d1 = q1 - zz1;
      local += d0 * d0 + d1 * d1;
      #pragma unroll
      for (int oc = 0; oc < 4; ++oc)          // fused post 1x1 conv (q_st == q)
        g[(((size_t)nn * 4 + oc) << 12) + yx] =
            postw[oc * 2] * q0 + postw[oc * 2 + 1] * q1 + postb[oc];
    }
  }
  if (m == 0) atomicAdd(vq_sum, local);
}

// deconv1 (gather form of lhs-dilated conv): [128,4,64,64] -> [128,16,128,128] + stats
__global__ void k_deconv1(const float* __restrict__ gin, const float* __restrict__ w,
                          const float* __restrict__ b, float* __restrict__ out,
                          float* ssum, float* ssq) {
  __shared__ float sw[1024];
  __shared__ float sb[16];
  int tid = threadIdx.x;
  for (int i = tid; i < 1024; i += 256) sw[i] = w[i];
  if (tid < 16) sb[tid] = b[tid];
  __syncthreads();
  int t = blockIdx.x * 256 + tid;           // 2,097,152 threads
  int n = t >> 14, yx = t & 16383;
  int oy = yx >> 7, ox = yx & 127;
  float acc[16];
  #pragma unroll
  for (int oc = 0; oc < 16; ++oc) acc[oc] = sb[oc];
  const float* base = gin + ((size_t)n << 14);
  int py = oy & 1, px = ox & 1;
  #pragma unroll
  for (int jy = 0; jy < 2; ++jy) {
    int ky = py + 2 * jy;
    int iy = (oy + ky - 2) >> 1;            // even by parity; >>1 ok for negatives
    if (iy < 0 || iy >= 64) continue;
    #pragma unroll
    for (int jx = 0; jx < 2; ++jx) {
      int kx = px + 2 * jx;
      int ix = (ox + kx - 2) >> 1;
      if (ix < 0 || ix >= 64) continue;
      #pragma unroll
      for (int ic = 0; ic < 4; ++ic) {
        float v = base[(ic << 12) + iy * 64 + ix];
        #pragma unroll
        for (int oc = 0; oc < 16; ++oc)
          acc[oc] += v * sw[((oc * 4 + ic) * 4 + ky) * 4 + kx];
      }
    }
  }
  #pragma unroll
  for (int oc = 0; oc < 16; ++oc) {
    out[(((size_t)n * 16 + oc) << 14) + yx] = acc[oc];
    float sA = wave_sum(acc[oc]), sQ = wave_sum(acc[oc] * acc[oc]);
    if ((tid & 31) == 0) { atomicAdd(&ssum[oc], sA); atomicAdd(&ssq[oc], sQ); }
  }
}

// deconv2 (fused bn3+relu reads) + tanh + recon loss: -> d_out [128,1,256,256]
__global__ void k_deconv2(const float* __restrict__ h3, const float* __restrict__ w,
                          const float* __restrict__ b, const float* sc3, const float* sh3,
                          const float* __restrict__ x, float* __restrict__ out,
                          float* recon_sum) {
  __shared__ float sw[256];
  __shared__ float ssc[16], ssh[16], sb1;
  int tid = threadIdx.x;
  sw[tid] = w[tid];
  if (tid == 0) sb1 = b[0];
  if (tid < 16) { ssc[tid] = sc3[tid]; ssh[tid] = sh3[tid]; }
  __syncthreads();
  int t = blockIdx.x * 256 + tid;           // 8,388,608 threads
  int n = t >> 16, yx = t & 65535;
  int oy = yx >> 8, ox = yx & 255;
  float acc = sb1;
  const float* base = h3 + ((size_t)n << 18);
  int py = oy & 1, px = ox & 1;
  #pragma unroll
  for (int jy = 0; jy < 2; ++jy) {
    int ky = py + 2 * jy;
    int iy = (oy + ky - 2) >> 1;
    if (iy < 0 || iy >= 128) continue;
    #pragma unroll
    for (int jx = 0; jx < 2; ++jx) {
      int kx = px + 2 * jx;
      int ix = (ox + kx - 2) >> 1;
      if (ix < 0 || ix >= 128) continue;
      #pragma unroll
      for (int ic = 0; ic < 16; ++ic) {
        float v = fmaxf(0.f, base[(ic << 14) + iy * 128 + ix] * ssc[ic] + ssh[ic]);
        acc += v * sw[(ic * 4 + ky) * 4 + kx];
      }
    }
  }
  float o = tanhf(acc);
  out[t] = o;
  float d = o - x[t];
  float s = wave_sum(d * d);
  if ((tid & 31) == 0) atomicAdd(recon_sum, s);
}

__global__ void k_losses(const float* vq_sum, const float* recon_sum, float* out) {
  if (threadIdx.x == 0) {
    float recon = recon_sum[0] * (1.f / 8388608.f);
    float cb = vq_sum[0] * (1.f / 1048576.f);   // codebook == commitment in fwd value
    out[8388608] = recon + cb + BETA_VQ * cb;   // total
    out[8388609] = recon;
    out[8388610] = cb;
    out[8388611] = cb;
  }
}

extern "C" void kernel_launch(void* const* d_in, const int* in_sizes, int n_in,
                              void* d_out, int out_size, void* d_ws, size_t ws_size,
                              hipStream_t stream) {
  (void)in_sizes; (void)n_in; (void)out_size; (void)ws_size;
  const float* x    = (const float*)d_in[0];
  const float* ew1  = (const float*)d_in[1];
  const float* eb1  = (const float*)d_in[2];
  const float* g1   = (const float*)d_in[3];
  const float* bb1  = (const float*)d_in[4];
  const float* ew2  = (const float*)d_in[5];
  const float* eb2  = (const float*)d_in[6];
  const float* g2   = (const float*)d_in[7];
  const float* bb2  = (const float*)d_in[8];
  const float* prw  = (const float*)d_in[9];
  const float* prb  = (const float*)d_in[10];
  const float* emb  = (const float*)d_in[11];
  const float* pow_ = (const float*)d_in[12];
  const float* pob  = (const float*)d_in[13];
  const float* dw1  = (const float*)d_in[14];
  const float* db1  = (const float*)d_in[15];
  const float* g3   = (const float*)d_in[16];
  const float* bb3  = (const float*)d_in[17];
  const float* dw2  = (const float*)d_in[18];
  const float* db2  = (const float*)d_in[19];
  float* out = (float*)d_out;
  float* ws = (float*)d_ws;

  float* buf1 = ws;                  // 33,554,432 f  [128,16,128,128]
  float* buf2 = ws + 33554432;       //  2,097,152 f  [128,4,64,64]
  float* zbuf = ws + 35651584;       //  1,048,576 f  [128,2,64,64]
  float* gbuf = ws + 36700160;       //  2,097,152 f  [128,4,64,64]
  float* st   = ws + 38797312;       // stats/loss block (146 f)
  float* s1sum = st;       float* s1sq = st + 16;
  float* s2sum = st + 32;  float* s2sq = st + 36;
  float* s3sum = st + 40;  float* s3sq = st + 56;
  float* sc1 = st + 72;  float* sh1 = st + 88;
  float* sc2 = st + 104; float* sh2 = st + 108;
  float* sc3 = st + 112; float* sh3 = st + 128;
  float* vqs = st + 144; float* rcs = st + 145;
  float* buf3 = buf1;                // reuse: buf1 dead after k_conv2

  k_zero<<<1, 256, 0, stream>>>(st, 146);
  k_conv1<<<8192, 256, 0, stream>>>(x, ew1, eb1, buf1, s1sum, s1sq);
  k_bnfold<<<1, 32, 0, stream>>>(s1sum, s1sq, g1, bb1, sc1, sh1, 16, 1.f / 2097152.f);
  k_conv2<<<2048, 256, 0, stream>>>(buf1, ew2, eb2, sc1, sh1, buf2, s2sum, s2sq);
  k_bnfold<<<1, 32, 0, stream>>>(s2sum, s2sq, g2, bb2, sc2, sh2, 4, 1.f / 524288.f);
  k_pre<<<2048, 256, 0, stream>>>(buf2, prw, prb, sc2, sh2, zbuf);
  k_vq<<<4096, 256, 0, stream>>>(zbuf, emb, pow_, pob, gbuf, vqs);
  k_deconv1<<<8192, 256, 0, stream>>>(gbuf, dw1, db1, buf3, s3sum, s3sq);
  k_bnfold<<<1, 32, 0, stream>>>(s3sum, s3sq, g3, bb3, sc3, sh3, 16, 1.f / 2097152.f);
  k_deconv2<<<32768, 256, 0, stream>>>(buf3, dw2, db2, sc3, sh3, x, out, rcs);
  k_losses<<<1, 1, 0, stream>>>(vqs, rcs, out);
  (void)db2; // dec_b2 folded via b arg above
}